// GraphPropagation_4260607558103
// MI455X (gfx1250) — compile-verified
//
#include <hip/hip_runtime.h>

// ---------------------------------------------------------------------------
// Graph propagation (excite/inhibit/normalize/activation-spread), CDNA5 gfx1250
//
// Structure: per iteration, kernel1 (excite: h1 = h + A_e h / ns_e) then a
// device-wide sync (separate launch) and kernel2 (inhibit on updated h1,
// relu+L2norm, activation spread). 16 receivers per 256-thread block; each
// thread owns 4 features (float4 gathers -> 256B coalesced row reads, L2
// resident since h is only 8MB vs 192MB L2). Zero-weight edges are skipped
// (activation starts one-hot, so early iterations gather almost nothing).
//
// The L2-norm uses V_WMMA_F32_16X16X4_F32: per 16-receiver tile, row-sums of
// the squared matrix T^2 (16x64) = T^2 x ones, accumulated over 16 K=4 chunks
// in exact f32. All lanes active (EXEC all-ones WMMA requirement).
// ---------------------------------------------------------------------------

#define BB     2
#define NN     16384
#define DD     64
#define DEG    64
#define ITERS  15
#define BN     (BB * NN)
#define THRESH 0.5f
#define EPS    1e-8f

typedef __attribute__((ext_vector_type(2))) float v2f;
typedef __attribute__((ext_vector_type(8))) float v8f;

// -------------------------- kernel 1: excite -------------------------------
__global__ __launch_bounds__(256) void gp_excite(
    const float* __restrict__ h_src,   // [BN*D]
    const float* __restrict__ act_src, // [BN]
    const float* __restrict__ val,     // [BN*DEG]
    const int*   __restrict__ cols,    // [BN*DEG]
    float*       __restrict__ h1)      // [BN*D]
{
    __shared__ int   scol[16 * DEG];
    __shared__ float sw  [16 * DEG];
    __shared__ float psum[16][16];

    const int tid    = threadIdx.x;
    const int g      = tid >> 4;        // local receiver 0..15
    const int lane16 = tid & 15;
    const int f0     = lane16 * 4;      // feature quad
    const int r0     = blockIdx.x * 16;
    const int r      = r0 + g;

    const size_t ebase = (size_t)r0 * DEG;       // 1024 edges for this block
    const int4   c4 = ((const int4*)(cols + ebase))[tid];
    const float4 v4 = ((const float4*)(val + ebase))[tid];

    const float a0 = act_src[c4.x], a1 = act_src[c4.y];
    const float a2 = act_src[c4.z], a3 = act_src[c4.w];
    const float w0 = (v4.x > THRESH ? v4.x : 0.0f) * a0;
    const float w1 = (v4.y > THRESH ? v4.y : 0.0f) * a1;
    const float w2 = (v4.z > THRESH ? v4.z : 0.0f) * a2;
    const float w3 = (v4.w > THRESH ? v4.w : 0.0f) * a3;

    const int eloc = tid * 4;                    // edges tid*4..tid*4+3, group g
    scol[eloc + 0] = c4.x; scol[eloc + 1] = c4.y;
    scol[eloc + 2] = c4.z; scol[eloc + 3] = c4.w;
    sw[eloc + 0] = w0; sw[eloc + 1] = w1; sw[eloc + 2] = w2; sw[eloc + 3] = w3;
    psum[g][lane16] = w0 + w1 + w2 + w3;
    __syncthreads();

    float ns = 0.0f;
#pragma unroll
    for (int i = 0; i < 16; ++i) ns += psum[g][i];
    ns = fmaxf(ns, 1.0f);
    const float inv = 1.0f / ns;

    float ax = 0.f, ay = 0.f, az = 0.f, aw = 0.f;
    const int eg = g * DEG;
#pragma unroll 4
    for (int e = 0; e < DEG; ++e) {
        const float w = sw[eg + e];
        if (w != 0.0f) {
            const float4 hv =
                *(const float4*)(h_src + (size_t)scol[eg + e] * DD + f0);
            ax += w * hv.x; ay += w * hv.y; az += w * hv.z; aw += w * hv.w;
        }
    }

    const size_t o = (size_t)r * DD + f0;
    const float4 hs = *(const float4*)(h_src + o);
    float4 out;
    out.x = hs.x + ax * inv; out.y = hs.y + ay * inv;
    out.z = hs.z + az * inv; out.w = hs.w + aw * inv;
    *(float4*)(h1 + o) = out;
}

// ------------------- kernel 2: inhibit + relu/L2norm + act -----------------
__global__ __launch_bounds__(256) void gp_inhibit(
    const float* __restrict__ h1,       // [BN*D] (globally updated by excite)
    const float* __restrict__ act_src,  // [BN]
    const float* __restrict__ val,
    const int*   __restrict__ cols,
    float*       __restrict__ out_slice,// [BN*D] preds[iter]
    float*       __restrict__ h_next,   // [BN*D] h for next iteration
    float*       __restrict__ act_dst)  // [BN]
{
    __shared__ int   scol[16 * DEG];
    __shared__ float sw  [16 * DEG];
    __shared__ float psum[16][16];
    __shared__ float pact[16][16];
    __shared__ float t2s [16][DD + 1];  // squared relu'd values (+1 pad)
    __shared__ float nrm [16];          // squared L2 norms per receiver

    const int tid    = threadIdx.x;
    const int g      = tid >> 4;
    const int lane16 = tid & 15;
    const int f0     = lane16 * 4;
    const int r0     = blockIdx.x * 16;
    const int r      = r0 + g;

    const size_t ebase = (size_t)r0 * DEG;
    const int4   c4 = ((const int4*)(cols + ebase))[tid];
    const float4 v4 = ((const float4*)(val + ebase))[tid];

    const float a0 = act_src[c4.x], a1 = act_src[c4.y];
    const float a2 = act_src[c4.z], a3 = act_src[c4.w];
    const float i0 = 1.0f - v4.x, i1 = 1.0f - v4.y;
    const float i2 = 1.0f - v4.z, i3 = 1.0f - v4.w;
    const float w0 = (i0 > THRESH ? i0 : 0.0f) * a0;
    const float w1 = (i1 > THRESH ? i1 : 0.0f) * a1;
    const float w2 = (i2 > THRESH ? i2 : 0.0f) * a2;
    const float w3 = (i3 > THRESH ? i3 : 0.0f) * a3;

    const int eloc = tid * 4;
    scol[eloc + 0] = c4.x; scol[eloc + 1] = c4.y;
    scol[eloc + 2] = c4.z; scol[eloc + 3] = c4.w;
    sw[eloc + 0] = w0; sw[eloc + 1] = w1; sw[eloc + 2] = w2; sw[eloc + 3] = w3;
    psum[g][lane16] = w0 + w1 + w2 + w3;
    pact[g][lane16] = fmaxf(fmaxf(a0, a1), fmaxf(a2, a3));
    __syncthreads();

    float ns = 0.0f, am = 0.0f;
#pragma unroll
    for (int i = 0; i < 16; ++i) { ns += psum[g][i]; am = fmaxf(am, pact[g][i]); }
    ns = fmaxf(ns, 1.0f);
    const float inv  = 1.0f / ns;
    const float recv = (am > 0.5f) ? 1.0f : 0.0f;

    float ax = 0.f, ay = 0.f, az = 0.f, aw = 0.f;
    const int eg = g * DEG;
#pragma unroll 4
    for (int e = 0; e < DEG; ++e) {
        const float w = sw[eg + e];
        if (w != 0.0f) {
            const float4 hv =
                *(const float4*)(h1 + (size_t)scol[eg + e] * DD + f0);
            ax += w * hv.x; ay += w * hv.y; az += w * hv.z; aw += w * hv.w;
        }
    }

    const size_t o = (size_t)r * DD + f0;
    const float4 hs = *(const float4*)(h1 + o);
    // relu(h1 - i_eff) + eps   (matches reference _relu_norm)
    const float x0 = fmaxf(hs.x - ax * inv, 0.0f) + EPS;
    const float x1 = fmaxf(hs.y - ay * inv, 0.0f) + EPS;
    const float x2 = fmaxf(hs.z - az * inv, 0.0f) + EPS;
    const float x3 = fmaxf(hs.w - aw * inv, 0.0f) + EPS;
    t2s[g][f0 + 0] = x0 * x0; t2s[g][f0 + 1] = x1 * x1;
    t2s[g][f0 + 2] = x2 * x2; t2s[g][f0 + 3] = x3 * x3;
    __syncthreads();

    // --- squared-norm row-sums via V_WMMA_F32_16X16X4_F32: D = T^2 x ones ---
    // A (16x4 f32) layout: VGPR0 holds K=0 (lanes 0-15) / K=2 (lanes 16-31),
    // VGPR1 holds K=1 / K=3. B = all ones, so every column of D is the row sum.
    {
        const int L   = tid & 31;
        const int M   = L & 15;
        const int kh  = (L & 16) ? 2 : 0;
        v2f ones; ones.x = 1.0f; ones.y = 1.0f;
        v8f gacc = {0.f, 0.f, 0.f, 0.f, 0.f, 0.f, 0.f, 0.f};
#pragma unroll
        for (int kc = 0; kc < 16; ++kc) {
            const int k0 = kc * 4;
            v2f a;
            a.x = t2s[M][k0 + kh];
            a.y = t2s[M][k0 + kh + 1];
            gacc = __builtin_amdgcn_wmma_f32_16x16x4_f32(
                false, a, false, ones, (short)0, gacc, false, false);
        }
        // D layout: VGPR i -> lanes 0-15: M=i, lanes 16-31: M=i+8 (all N equal)
        if (tid == 0) {
#pragma unroll
            for (int i = 0; i < 8; ++i) nrm[i] = gacc[i];
        }
        if (tid == 16) {
#pragma unroll
            for (int i = 0; i < 8; ++i) nrm[8 + i] = gacc[i];
        }
    }
    __syncthreads();

    const float n     = sqrtf(nrm[g]);
    const float scale = 1.0f / fmaxf(n, EPS);
    float4 out;
    out.x = x0 * scale; out.y = x1 * scale;
    out.z = x2 * scale; out.w = x3 * scale;
    *(float4*)(out_slice + o) = out;
    *(float4*)(h_next + o)    = out;

    if (lane16 == 0)  // one writer per receiver: act_new = clip(act+recv,0,1)
        act_dst[r] = fminf(act_src[r] + recv, 1.0f);
}

// ------------------------------ host side ----------------------------------
extern "C" void kernel_launch(void* const* d_in, const int* in_sizes, int n_in,
                              void* d_out, int out_size, void* d_ws, size_t ws_size,
                              hipStream_t stream) {
    (void)in_sizes; (void)n_in; (void)out_size; (void)ws_size;
    const float* h0        = (const float*)d_in[0];
    const float* val       = (const float*)d_in[1];
    const float* activated = (const float*)d_in[2];
    /* d_in[3] = rows: implicit (rows[e] = e / DEG), unused */
    const int*   cols      = (const int*)d_in[4];
    float* out = (float*)d_out;

    float* ws   = (float*)d_ws;
    float* h1   = ws;                          // BN*D
    float* hcur = h1 + (size_t)BN * DD;        // BN*D
    float* actA = hcur + (size_t)BN * DD;      // BN
    float* actB = actA + BN;                   // BN

    const dim3 grid(BN / 16), block(256);
    for (int it = 0; it < ITERS; ++it) {
        const float* hsrc = (it == 0) ? h0 : hcur;
        const float* asrc = (it == 0) ? activated : ((it & 1) ? actA : actB);
        float*       adst = (it & 1) ? actB : actA;

        gp_excite<<<grid, block, 0, stream>>>(hsrc, asrc, val, cols, h1);
        gp_inhibit<<<grid, block, 0, stream>>>(h1, asrc, val, cols,
                                               out + (size_t)it * BN * DD,
                                               hcur, adst);
    }
}